// TDA_NegCache_49357764165817
// MI455X (gfx1250) — compile-verified
//
#include <hip/hip_runtime.h>
#include <math.h>

#define N_SAMPLES 8192
#define K_CLASSES 1000
#define S_SLOTS   8
#define D_DIM     1024

typedef __attribute__((ext_vector_type(2))) float v2f;
typedef __attribute__((ext_vector_type(8))) float v8f;

// ---------------------------------------------------------------------------
// Kernel 1: per-sample softmax stats: argmax label, entropy H, gate flag.
// One 256-thread block per sample, three LDS reduction passes over K=1000.
// ---------------------------------------------------------------------------
__global__ void __launch_bounds__(256)
stats_kernel(const float* __restrict__ logits,
             int* __restrict__ label, float* __restrict__ Harr,
             int* __restrict__ gate) {
  __shared__ float sv[256];
  __shared__ int   si[256];
  const int n = blockIdx.x;
  const float* L = logits + (size_t)n * K_CLASSES;
  const int t = threadIdx.x;

  // pass 1: max + first-occurrence argmax
  float m = -INFINITY; int mi = 0;
  for (int j = t; j < K_CLASSES; j += 256) {
    float v = L[j];
    if (v > m) { m = v; mi = j; }
  }
  sv[t] = m; si[t] = mi;
  __syncthreads();
  for (int s = 128; s > 0; s >>= 1) {
    if (t < s) {
      float ov = sv[t + s]; int oi = si[t + s];
      if (ov > sv[t] || (ov == sv[t] && oi < si[t])) { sv[t] = ov; si[t] = oi; }
    }
    __syncthreads();
  }
  m = sv[0]; mi = si[0];
  __syncthreads();

  // pass 2: partition function Z
  float z = 0.f;
  for (int j = t; j < K_CLASSES; j += 256) z += expf(L[j] - m);
  sv[t] = z; __syncthreads();
  for (int s = 128; s > 0; s >>= 1) { if (t < s) sv[t] += sv[t + s]; __syncthreads(); }
  const float Z = sv[0];
  const float invZ = 1.0f / Z;
  __syncthreads();

  // pass 3: entropy  H = -sum p*log(p + 1e-6)
  float hs = 0.f;
  for (int j = t; j < K_CLASSES; j += 256) {
    float p = expf(L[j] - m) * invZ;
    hs += p * logf(p + 1e-6f);
  }
  sv[t] = hs; __syncthreads();
  for (int s = 128; s > 0; s >>= 1) { if (t < s) sv[t] += sv[t + s]; __syncthreads(); }

  if (t == 0) {
    float H = -sv[0];
    label[n] = mi;
    Harr[n]  = H;
    // p[label] = exp(m - m)/Z = 1/Z
    gate[n] = (invZ > 0.03f && H > 0.2f && H < 0.5f) ? 1 : 0;
  }
}

// ---------------------------------------------------------------------------
// Kernel 2: sequential entropy-gated scan, one thread per class.
// Updates within a class are the only order-dependent part; a slot's final
// content is the LAST sample written, so we only track (entropy, src index).
// ---------------------------------------------------------------------------
__global__ void __launch_bounds__(256)
scan_kernel(const float* __restrict__ ent_in,
            const int* __restrict__ label,
            const float* __restrict__ Harr,
            const int* __restrict__ gate,
            int* __restrict__ src) {
  const int c = blockIdx.x * blockDim.x + threadIdx.x;
  if (c >= K_CLASSES) return;

  float ent[S_SLOTS];
  int   sr[S_SLOTS];
#pragma unroll
  for (int s = 0; s < S_SLOTS; ++s) {
    ent[s] = ent_in[(size_t)c * S_SLOTS + s];
    sr[s]  = -1;
  }

  for (int i = 0; i < N_SAMPLES; ++i) {
    if (label[i] != c) continue;     // uniform loads, lane-divergent filter
    if (!gate[i]) continue;
    float h = Harr[i];
    // evict slot = first argmax of entropies (matches jnp.argmax tie rule)
    float mx = ent[0]; int midx = 0;
#pragma unroll
    for (int s = 1; s < S_SLOTS; ++s)
      if (ent[s] > mx) { mx = ent[s]; midx = s; }
    // cond ... & any(H < row)  <=>  H < max(row)
    if (h < mx) { ent[midx] = h; sr[midx] = i; }
  }

#pragma unroll
  for (int s = 0; s < S_SLOTS; ++s) src[(size_t)c * S_SLOTS + s] = sr[s];
}

// ---------------------------------------------------------------------------
// Kernel 3: materialize final memory [K*S, D]: row = x[src] or original mem.
// One block per (k,s) row; 256 threads x float4 = 1024 floats.
// ---------------------------------------------------------------------------
__global__ void __launch_bounds__(256)
gather_kernel(const float* __restrict__ x,
              const float* __restrict__ mem_in,
              const int* __restrict__ src,
              float* __restrict__ mem_ws) {
  const int b = blockIdx.x;
  const int s = src[b];
  const float* sp = (s >= 0) ? (x + (size_t)s * D_DIM)
                             : (mem_in + (size_t)b * D_DIM);
  const float4* sp4 = (const float4*)sp;
  float4* dp4 = (float4*)(mem_ws + (size_t)b * D_DIM);
  dp4[threadIdx.x] = sp4[threadIdx.x];
}

// ---------------------------------------------------------------------------
// one xor-butterfly step via ds_swizzle group-of-32
// (and=0x1f, or=0, xor=1/2/4 -> offsets 0x041F / 0x081F / 0x101F)
// CODE is a template parameter: the builtin requires an immediate constant.
// ---------------------------------------------------------------------------
template <int CODE>
__device__ __forceinline__ float swz_add(float e) {
  return e + __int_as_float(__builtin_amdgcn_ds_swizzle(__float_as_int(e), CODE));
}

// ---------------------------------------------------------------------------
// Kernel 4: fp32 WMMA GEMM + fused exp/slot-reduction epilogue.
//   A[n, ks] = dot(x[n,:], mem[ks,:]),  out[n,k] = -sum_{s} exp(A - 1)
// Register-blocked: each wave owns a 64(M) x 64(KS) tile = 4x4 WMMA subtiles
// (16 v8f accumulators). Per d-step of 4: 8 b64 loads feed 16 WMMAs
// -> 16 FLOP per cache byte. All 8 waves in a block share ks0, so B
// fragments broadcast from L0/L1. Epilogue is staged so the 8 ds_swizzle
// chains per subtile run concurrently instead of one serialized chain.
// ---------------------------------------------------------------------------
__global__ void __launch_bounds__(256)
gemm_kernel(const float* __restrict__ x,
            const float* __restrict__ memw,
            float* __restrict__ out) {
  const int lane   = threadIdx.x & 31;
  const int waveId = threadIdx.x >> 5;
  const int lid    = lane & 15;
  const int half   = lane >> 4;

  const int m0  = (blockIdx.y * 8 + waveId) * 64;  // sample rows (per wave)
  const int ks0 = blockIdx.x * 64;                 // flattened k*S+s columns

  // A 16x4 f32 layout: lanes 0-15 hold K=0,1 ; lanes 16-31 hold K=2,3
  const float* aBase = x    + (size_t)(m0  + lid) * D_DIM + 2 * half;
  // B 4x16 f32 layout is symmetric: per-lane column = ks row of mem
  const float* bBase = memw + (size_t)(ks0 + lid) * D_DIM + 2 * half;

  v8f c[4][4];
#pragma unroll
  for (int i = 0; i < 4; ++i)
#pragma unroll
    for (int j = 0; j < 4; ++j) c[i][j] = (v8f){};

#pragma unroll 2
  for (int d = 0; d < D_DIM; d += 4) {
    v2f a[4], b[4];
#pragma unroll
    for (int i = 0; i < 4; ++i)
      a[i] = *(const v2f*)(aBase + (size_t)(i * 16) * D_DIM + d);
#pragma unroll
    for (int j = 0; j < 4; ++j)
      b[j] = *(const v2f*)(bBase + (size_t)(j * 16) * D_DIM + d);
#pragma unroll
    for (int i = 0; i < 4; ++i)
#pragma unroll
      for (int j = 0; j < 4; ++j)
        c[i][j] = __builtin_amdgcn_wmma_f32_16x16x4_f32(
            false, a[i], false, b[j], (short)0, c[i][j], false, false);
  }

  // C/D layout: VGPR v, lane l -> M = v + 8*(l>>4), N = l&15 (N = ks offset)
  const bool writer = ((lane & 7) == 0);
#pragma unroll
  for (int i = 0; i < 4; ++i) {
#pragma unroll
    for (int j = 0; j < 4; ++j) {
      float e[8];
      // stage 0: 8 independent exps
#pragma unroll
      for (int v = 0; v < 8; ++v) e[v] = __expf(c[i][j][v] - 1.0f);
      // stages 1..3: xor-butterfly over lane bits 0..2; 8 concurrent DS chains
#pragma unroll
      for (int v = 0; v < 8; ++v) e[v] = swz_add<0x041F>(e[v]);
#pragma unroll
      for (int v = 0; v < 8; ++v) e[v] = swz_add<0x081F>(e[v]);
#pragma unroll
      for (int v = 0; v < 8; ++v) e[v] = swz_add<0x101F>(e[v]);
      if (writer) {
        const int k = (ks0 + j * 16 + lid) >> 3;   // S_SLOTS == 8
        const int nb = m0 + i * 16 + 8 * half;
#pragma unroll
        for (int v = 0; v < 8; ++v)
          out[(size_t)(nb + v) * K_CLASSES + k] = -e[v];
      }
    }
  }
}

// ---------------------------------------------------------------------------
extern "C" void kernel_launch(void* const* d_in, const int* in_sizes, int n_in,
                              void* d_out, int out_size, void* d_ws, size_t ws_size,
                              hipStream_t stream) {
  const float* x      = (const float*)d_in[0];  // [N, D]
  const float* tlog   = (const float*)d_in[1];  // [N, K]
  const float* mem_in = (const float*)d_in[2];  // [K, S, D]
  const float* ent_in = (const float*)d_in[3];  // [K, S]
  // d_in[4] = memory_state (unused; not part of output)
  float* out = (float*)d_out;                   // [N, K]

  float* mem_ws = (float*)d_ws;
  const size_t memElems = (size_t)K_CLASSES * S_SLOTS * D_DIM;
  int*   label = (int*)(mem_ws + memElems);
  float* Harr  = (float*)(label + N_SAMPLES);
  int*   gate  = (int*)(Harr + N_SAMPLES);
  int*   src   = (int*)(gate + N_SAMPLES);

  stats_kernel<<<N_SAMPLES, 256, 0, stream>>>(tlog, label, Harr, gate);
  scan_kernel<<<(K_CLASSES + 255) / 256, 256, 0, stream>>>(ent_in, label, Harr,
                                                           gate, src);
  gather_kernel<<<K_CLASSES * S_SLOTS, 256, 0, stream>>>(x, mem_in, src, mem_ws);

  // block = 8 waves, each wave a 64x64 tile -> block covers 512(M) x 64(KS)
  dim3 grid((K_CLASSES * S_SLOTS) / 64, N_SAMPLES / 512);  // (125, 16)
  gemm_kernel<<<grid, 256, 0, stream>>>(x, mem_ws, out);
}